// ShortConvolution_58944131170395
// MI455X (gfx1250) — compile-verified
//
#include <hip/hip_runtime.h>

// Problem constants (from the reference setup)
#define BATCH 4
#define SEQ   4096
#define DMOD  2048
#define KSZ   4
#define LTILE 8              // l-positions per thread
#define D4    (DMOD / 4)     // float4 groups along d

typedef float v4f __attribute__((ext_vector_type(4)));

// SiLU with single-instruction reciprocal (v_rcp_f32) instead of IEEE divide:
// cuts ~8 VALU ops per element, keeping the kernel HBM-bound, not VALU-bound.
__device__ __forceinline__ float silu_fast(float v) {
    float e = __expf(-v);                       // v_mul (log2e) + v_exp_f32
    return v * __builtin_amdgcn_rcpf(1.0f + e); // v_add + v_rcp_f32 + v_mul
}

__global__ __launch_bounds__(256)
void ShortConvolution_58944131170395_kernel(const float* __restrict__ x,
                                            const float* __restrict__ w,
                                            float* __restrict__ out)
{
    // Flat thread index -> (b, l-tile, d4). d4 varies fastest => coalesced 128-bit
    // accesses across the wave32 along the contiguous d axis.
    const int idx = blockIdx.x * blockDim.x + threadIdx.x;
    const int d4  = idx & (D4 - 1);            // D4 = 512
    const int t2  = idx >> 9;
    const int lt  = t2 & ((SEQ / LTILE) - 1);  // 512 tiles (uniform per block)
    const int b   = t2 >> 9;

    const int    l0   = lt * LTILE;
    const size_t base = ((size_t)b * SEQ + (size_t)l0) * DMOD + (size_t)d4 * 4;
    const float* xp = x + base;
    float*       op = out + base;

    // Per-channel taps: weight layout (D, 1, K) -> K=4 contiguous floats / channel.
    const v4f* wp = (const v4f*)w + (size_t)d4 * 4;
    const v4f w0 = wp[0], w1 = wp[1], w2 = wp[2], w3 = wp[3];

    // Register window: rows l0-3 .. l0+LTILE-1, all loads issued back-to-back
    // (deep MLP before the first s_wait_loadcnt).
    v4f xs[LTILE + 3];
    if (l0 >= KSZ - 1) {
#pragma unroll
        for (int i = 0; i < LTILE + 3; ++i)
            xs[i] = *(const v4f*)(xp + ((long)i - 3) * DMOD);
    } else {
        const v4f z = (v4f)(0.0f);
        xs[0] = z; xs[1] = z; xs[2] = z;   // causal zero padding at l<0
#pragma unroll
        for (int i = 0; i < LTILE; ++i)
            xs[i + 3] = *(const v4f*)(xp + (long)i * DMOD);
    }

    // Speculative prefetch of the next l-tile of this (b, d4) stream into L2
    // (gfx1250 global_prefetch_b8; translation faults are silently dropped).
    __builtin_prefetch(xp + (size_t)LTILE * DMOD, 0, 1);

#pragma unroll
    for (int t = 0; t < LTILE; ++t) {
        const v4f a  = xs[t];       // x[l-3]
        const v4f bb = xs[t + 1];   // x[l-2]
        const v4f c  = xs[t + 2];   // x[l-1]
        const v4f d  = xs[t + 3];   // x[l]
        v4f y;
        y.x = fmaf(w0.w, d.x, fmaf(w0.z, c.x, fmaf(w0.y, bb.x, w0.x * a.x)));
        y.y = fmaf(w1.w, d.y, fmaf(w1.z, c.y, fmaf(w1.y, bb.y, w1.x * a.y)));
        y.z = fmaf(w2.w, d.z, fmaf(w2.z, c.z, fmaf(w2.y, bb.z, w2.x * a.z)));
        y.w = fmaf(w3.w, d.w, fmaf(w3.z, c.w, fmaf(w3.y, bb.w, w3.x * a.w)));
        y.x = silu_fast(y.x);
        y.y = silu_fast(y.y);
        y.z = silu_fast(y.z);
        y.w = silu_fast(y.w);
        // Output is write-once / never re-read: non-temporal 128-bit store.
        __builtin_nontemporal_store(y, (v4f*)(op + (long)t * DMOD));
    }
}

extern "C" void kernel_launch(void* const* d_in, const int* in_sizes, int n_in,
                              void* d_out, int out_size, void* d_ws, size_t ws_size,
                              hipStream_t stream)
{
    const float* x = (const float*)d_in[0];   // (4, 4096, 2048) fp32
    const float* w = (const float*)d_in[1];   // (2048, 1, 4)    fp32
    float* out = (float*)d_out;               // (4, 4096, 2048) fp32

    const int total_threads = BATCH * (SEQ / LTILE) * D4;  // 1,048,576
    const int block = 256;                                  // 8 wave32s
    ShortConvolution_58944131170395_kernel<<<total_threads / block, block, 0, stream>>>(x, w, out);
}